// Pool_17712445128830
// MI455X (gfx1250) — compile-verified
//
#include <hip/hip_runtime.h>
#include <math.h>

#define TOTAL_TOKENS 131072
#define D_MODEL 1024
#define NUM_SEGS 64

typedef __attribute__((ext_vector_type(2))) float v2f;
typedef __attribute__((ext_vector_type(8))) float v8f;

// ---------------------------------------------------------------------------
// Kernel 0: zero the output (harness poisons d_out; pool pass uses atomics).
// ---------------------------------------------------------------------------
__global__ void zero_out_kernel(float* __restrict__ out, int n) {
    int i = blockIdx.x * blockDim.x + threadIdx.x;
    if (i < n) out[i] = 0.0f;
}

// ---------------------------------------------------------------------------
// Kernel 1: scores[t] = dot(x[t,:], W) + b  via V_WMMA_F32_16X16X4_F32.
//
// Each wave32 computes 16 token scores. Mapping (D = A x B + C):
//   A[m,k] = W[kb+k]  replicated over rows m   (16x4, f32 A layout:
//            vgpr0: lanes0-15 K=0, lanes16-31 K=2; vgpr1: K=1 / K=3)
//   B[k,n] = x[tokBase+n, kb+k]                (4x16, mirrored split)
//   => D[m,n] = score contribution of token n, identical for every m.
// Accumulate C over kb = 0..1020 step 4; lane n reads its score from c[0].
// EXEC is all-ones through the whole WMMA loop (uniform control flow).
// ---------------------------------------------------------------------------
__global__ void score_wmma_kernel(const float* __restrict__ x,
                                  const float* __restrict__ W,
                                  const float* __restrict__ bptr,
                                  float* __restrict__ scores) {
    __shared__ float wLDS[D_MODEL];
    for (int i = threadIdx.x; i < D_MODEL; i += blockDim.x)
        wLDS[i] = W[i];
    __syncthreads();

    const int lane    = threadIdx.x & 31;
    const int wave    = threadIdx.x >> 5;
    const int tokBase = blockIdx.x * 128 + wave * 16;   // 8 waves * 16 tokens
    const int n       = lane & 15;                      // column / token index
    const int koff    = (lane >= 16) ? 2 : 0;           // K split per lane half

    const float* xrow = x + (size_t)(tokBase + n) * D_MODEL + koff;

    v8f c = {};
    for (int kb = 0; kb < D_MODEL; kb += 4) {
        v2f a, bm;
        a.x = wLDS[kb + koff];
        a.y = wLDS[kb + koff + 1];
        const float2 xv = *reinterpret_cast<const float2*>(xrow + kb);
        bm.x = xv.x;
        bm.y = xv.y;
        // 8 args: (neg_a, A, neg_b, B, c_mod, C, reuse_a, reuse_b)
        c = __builtin_amdgcn_wmma_f32_16x16x4_f32(
                false, a, false, bm, (short)0, c, false, false);
    }

    if (lane < 16) {
        // c[0]: M=0 row, N=lane -> score of token tokBase+lane
        scores[tokBase + lane] = c[0] + bptr[0];
    }
}

// ---------------------------------------------------------------------------
// Kernel 2: per-segment max and sum(exp(score-max)). One block per segment;
// the score vector is only 512 KB total so this pass is negligible.
// ---------------------------------------------------------------------------
__global__ void segstat_kernel(const float* __restrict__ scores,
                               const int* __restrict__ cu,
                               float* __restrict__ seg_max,
                               float* __restrict__ seg_sum) {
    const int s   = blockIdx.x;
    const int beg = cu[s];
    const int end = cu[s + 1];
    __shared__ float red[256];

    float m = -3.402823466e38f;
    for (int t = beg + threadIdx.x; t < end; t += blockDim.x)
        m = fmaxf(m, scores[t]);
    red[threadIdx.x] = m;
    __syncthreads();
    for (int off = 128; off > 0; off >>= 1) {
        if (threadIdx.x < off)
            red[threadIdx.x] = fmaxf(red[threadIdx.x], red[threadIdx.x + off]);
        __syncthreads();
    }
    const float mx = red[0];
    __syncthreads();

    float sum = 0.0f;
    for (int t = beg + threadIdx.x; t < end; t += blockDim.x)
        sum += __expf(scores[t] - mx);
    red[threadIdx.x] = sum;
    __syncthreads();
    for (int off = 128; off > 0; off >>= 1) {
        if (threadIdx.x < off)
            red[threadIdx.x] += red[threadIdx.x + off];
        __syncthreads();
    }
    if (threadIdx.x == 0) {
        seg_max[s] = mx;
        seg_sum[s] = red[0];
    }
}

// ---------------------------------------------------------------------------
// Kernel 3: out[s,:] += sum_t w_t * x[t,:]. Each block owns 256 consecutive
// tokens; thread j owns dims [4j..4j+3] (float4 reads are fully coalesced:
// one 4 KB row per token per block iteration). Segment boundaries inside a
// block are rare (64 boundaries / 512 blocks), flushed with f32 atomics.
// ---------------------------------------------------------------------------
__global__ void pool_kernel(const float* __restrict__ x,
                            const float* __restrict__ scores,
                            const int* __restrict__ cu,
                            const float* __restrict__ seg_max,
                            const float* __restrict__ seg_sum,
                            float* __restrict__ out) {
    __shared__ int   cuLDS[NUM_SEGS + 1];
    __shared__ float wv[256];
    __shared__ int   sg[256];

    if (threadIdx.x <= NUM_SEGS) cuLDS[threadIdx.x] = cu[threadIdx.x];
    __syncthreads();

    const int base = blockIdx.x * 256;
    {
        const int t = base + (int)threadIdx.x;
        int lo = 0, hi = NUM_SEGS;           // largest s with cu[s] <= t
        while (hi - lo > 1) {
            const int mid = (lo + hi) >> 1;
            if (cuLDS[mid] <= t) lo = mid; else hi = mid;
        }
        const int s = lo;
        wv[threadIdx.x] = __expf(scores[t] - seg_max[s]) / seg_sum[s];
        sg[threadIdx.x] = s;
    }
    __syncthreads();

    const int d0 = (int)threadIdx.x * 4;
    float4 acc = make_float4(0.0f, 0.0f, 0.0f, 0.0f);
    int cur = sg[0];
    for (int i = 0; i < 256; ++i) {
        const int s = sg[i];                 // uniform across the block
        if (s != cur) {
            float* o = out + (size_t)cur * D_MODEL + d0;
            atomicAdd(o + 0, acc.x);
            atomicAdd(o + 1, acc.y);
            atomicAdd(o + 2, acc.z);
            atomicAdd(o + 3, acc.w);
            acc = make_float4(0.0f, 0.0f, 0.0f, 0.0f);
            cur = s;
        }
        const float  wt = wv[i];
        const float4 xv = *reinterpret_cast<const float4*>(
                              x + (size_t)(base + i) * D_MODEL + d0);
        acc.x += wt * xv.x;
        acc.y += wt * xv.y;
        acc.z += wt * xv.z;
        acc.w += wt * xv.w;
    }
    float* o = out + (size_t)cur * D_MODEL + d0;
    atomicAdd(o + 0, acc.x);
    atomicAdd(o + 1, acc.y);
    atomicAdd(o + 2, acc.z);
    atomicAdd(o + 3, acc.w);
}

// ---------------------------------------------------------------------------
extern "C" void kernel_launch(void* const* d_in, const int* in_sizes, int n_in,
                              void* d_out, int out_size, void* d_ws, size_t ws_size,
                              hipStream_t stream) {
    (void)in_sizes; (void)n_in; (void)out_size; (void)ws_size;

    const float* x  = (const float*)d_in[0];
    const int*   cu = (const int*)d_in[1];
    const float* W  = (const float*)d_in[2];
    const float* b  = (const float*)d_in[3];
    float*       out = (float*)d_out;

    float* scores  = (float*)d_ws;                 // TOTAL_TOKENS floats
    float* seg_max = scores + TOTAL_TOKENS;        // NUM_SEGS floats
    float* seg_sum = seg_max + NUM_SEGS;           // NUM_SEGS floats

    zero_out_kernel<<<(NUM_SEGS * D_MODEL + 255) / 256, 256, 0, stream>>>(
        out, NUM_SEGS * D_MODEL);

    score_wmma_kernel<<<TOTAL_TOKENS / 128, 256, 0, stream>>>(
        x, W, b, scores);

    segstat_kernel<<<NUM_SEGS, 256, 0, stream>>>(
        scores, cu, seg_max, seg_sum);

    pool_kernel<<<TOTAL_TOKENS / 256, 256, 0, stream>>>(
        x, scores, cu, seg_max, seg_sum, out);
}